// IntegratedLoss_21096879357952
// MI455X (gfx1250) — compile-verified
//
#include <hip/hip_runtime.h>
#include <math.h>

// ---------------------------------------------------------------------------
// IntegratedLoss (rotated-box detection loss) for MI455X / gfx1250.
// Geometry is fp32 VALU-bound; final reductions use V_WMMA_F32_16X16X4_F32
// with an all-ones B matrix (exact fp32 sum, layout-agnostic).
// ---------------------------------------------------------------------------

constexpr int   Bsz  = 2;
constexpr int   Anum = 16384;
constexpr int   Mnum = 32;
constexpr int   Cnum = 15;
constexpr int   NBLK = 128;          // loss partial blocks per image (A / 128)
constexpr float MD_THRES = 0.5f;
constexpr float EPSV = 1e-8f;
constexpr float BETA = 1.0f / 9.0f;
constexpr float DEG2RAD = 0.017453292519943295f;

struct P2 { float x, y; };

__device__ __forceinline__ float cross2(float ax, float ay, float bx, float by) {
    return ax * by - ay * bx;
}

__device__ __forceinline__ void rcorners(const float* b, P2 c[4]) {
    const float lx[4] = {-0.5f, 0.5f, 0.5f, -0.5f};
    const float ly[4] = {-0.5f, -0.5f, 0.5f, 0.5f};
    float t = b[4] * DEG2RAD;
    float cs = cosf(t), sn = sinf(t);
    for (int k = 0; k < 4; ++k) {
        float dx = b[2] * lx[k], dy = b[3] * ly[k];
        c[k].x = b[0] + dx * cs - dy * sn;
        c[k].y = b[1] + dx * sn + dy * cs;
    }
}

__device__ bool pt_in_quad(float px, float py, const P2* v, const P2* e) {
    bool allp = true, alln = true;
    for (int j = 0; j < 4; ++j) {
        float cr = e[j].x * (py - v[j].y) - e[j].y * (px - v[j].x);
        allp = allp && (cr >= -1e-6f);
        alln = alln && (cr <= 1e-6f);
    }
    return allp || alln;
}

// Convex-polygon intersection area of two quads, mirroring the reference:
// candidate points = corners-in-other-quad + all 16 segment intersections,
// sorted by angle around centroid, shoelace.
__device__ float quad_inter(const P2* ca, const P2* cb) {
    P2 r[4], s[4];
    for (int k = 0; k < 4; ++k) {
        r[k].x = ca[(k + 1) & 3].x - ca[k].x;
        r[k].y = ca[(k + 1) & 3].y - ca[k].y;
        s[k].x = cb[(k + 1) & 3].x - cb[k].x;
        s[k].y = cb[(k + 1) & 3].y - cb[k].y;
    }
    P2 cand[24];
    bool mask[24];
    for (int k = 0; k < 4; ++k) {
        cand[k] = ca[k];     mask[k]     = pt_in_quad(ca[k].x, ca[k].y, cb, s);
        cand[4 + k] = cb[k]; mask[4 + k] = pt_in_quad(cb[k].x, cb[k].y, ca, r);
    }
    int q = 8;
    for (int i = 0; i < 4; ++i) {
        for (int j = 0; j < 4; ++j, ++q) {
            float rxs  = cross2(r[i].x, r[i].y, s[j].x, s[j].y);
            float qpx  = cb[j].x - ca[i].x, qpy = cb[j].y - ca[i].y;
            float qpxs = cross2(qpx, qpy, s[j].x, s[j].y);
            float qpxr = cross2(qpx, qpy, r[i].x, r[i].y);
            bool  ok   = fabsf(rxs) > 1e-10f;
            float den  = ok ? rxs : 1.0f;
            float t = qpxs / den, u = qpxr / den;
            mask[q] = ok && (t >= 0.f) && (t <= 1.f) && (u >= 0.f) && (u <= 1.f);
            cand[q].x = ca[i].x + t * r[i].x;
            cand[q].y = ca[i].y + t * r[i].y;
        }
    }
    int n = 0; float cx = 0.f, cy = 0.f;
    for (int k = 0; k < 24; ++k)
        if (mask[k]) { ++n; cx += cand[k].x; cy += cand[k].y; }
    float inv = 1.0f / (float)(n > 1 ? n : 1);
    cx *= inv; cy *= inv;
    float ang[24]; int ord[24];
    for (int k = 0; k < 24; ++k) {
        ang[k] = mask[k] ? atan2f(cand[k].y - cy, cand[k].x - cx) : 1e9f;
        ord[k] = k;
    }
    // stable insertion sort by angle (matches jnp.argsort stability)
    for (int i = 1; i < 24; ++i) {
        int oi = ord[i]; float key = ang[oi]; int j = i - 1;
        while (j >= 0 && ang[ord[j]] > key) { ord[j + 1] = ord[j]; --j; }
        ord[j + 1] = oi;
    }
    P2 sp[24];
    P2 first = cand[ord[0]];
    for (int k = 0; k < 24; ++k)
        sp[k] = mask[ord[k]] ? cand[ord[k]] : first;
    float area = 0.f;
    for (int k = 0; k < 24; ++k) {
        const P2& a0 = sp[k];
        const P2& b0 = sp[(k + 1) % 24];
        area += a0.x * b0.y - b0.x * a0.y;
    }
    area = 0.5f * fabsf(area);
    return (n >= 3) ? area : 0.f;
}

// rotated IoU gated by AABB (min_area_square) overlap >= 0.1, as in reference
__device__ float rbox_iou_pair(const float* bx, const float* gt) {
    float sa = 0.5f * fmaxf(bx[2], bx[3]);
    float sb = 0.5f * fmaxf(gt[2], gt[3]);
    float ax1 = bx[0] - sa, ay1 = bx[1] - sa, ax2 = bx[0] + sa, ay2 = bx[1] + sa;
    float bx1 = gt[0] - sb, by1 = gt[1] - sb, bx2 = gt[0] + sb, by2 = gt[1] + sb;
    float iw = fmaxf(fminf(ax2, bx2) - fmaxf(ax1, bx1), 0.f);
    float ih = fmaxf(fminf(ay2, by2) - fmaxf(ay1, by1), 0.f);
    float i0 = iw * ih;
    float ind = i0 / ((ax2 - ax1) * (ay2 - ay1) + (bx2 - bx1) * (by2 - by1) - i0 + EPSV);
    if (ind < 0.1f) return 0.f;
    P2 ca[4], cb[4];
    rcorners(bx, ca);
    rcorners(gt, cb);
    float inter = quad_inter(ca, cb);
    return inter / (bx[2] * bx[3] + gt[2] * gt[3] - inter + EPSV);
}

// ---- Stage 1: matching degree md[b][a][m] ----------------------------------
__global__ void k_md(const float* __restrict__ anc, const float* __restrict__ ranc,
                     const float* __restrict__ ann, float* __restrict__ md) {
    int idx = blockIdx.x * blockDim.x + threadIdx.x;
    if (idx >= Bsz * Anum * Mnum) return;
    int m = idx % Mnum;
    int a = (idx / Mnum) % Anum;
    int b = idx / (Mnum * Anum);
    const float* pa = anc  + ((size_t)b * Anum + a) * 5;
    const float* pr = ranc + ((size_t)b * Anum + a) * 5;
    const float* pg = ann  + ((size_t)b * Mnum + m) * 6;
    float ov_bf = rbox_iou_pair(pa, pg);
    float ov_af = rbox_iou_pair(pr, pg);
    // |BF*bf + AF*af - |af-bf|^VAR| with BF=1, AF=0.5, VAR=1
    md[idx] = fabsf(ov_bf + 0.5f * ov_af - fabsf(ov_af - ov_bf));
}

// ---- Stage 2: per-anchor max/argmax over M ---------------------------------
__global__ void k_anchor_max(const float* __restrict__ md,
                             float* __restrict__ iou_max, int* __restrict__ iou_arg) {
    int idx = blockIdx.x * blockDim.x + threadIdx.x;
    if (idx >= Bsz * Anum) return;
    const float* row = md + (size_t)idx * Mnum;
    float best = row[0]; int bi = 0;
    for (int m = 1; m < Mnum; ++m) {
        float v = row[m];
        if (v > best) { best = v; bi = m; }  // first-occurrence tie
    }
    iou_max[idx] = best;
    iou_arg[idx] = bi;
}

// ---- Stage 3: per-GT argmax over anchors (optionally pos-masked) -----------
__global__ void k_gt_max(const float* __restrict__ md, const int* __restrict__ pos,
                         int use_pos, float* __restrict__ outv, int* __restrict__ outi) {
    int b = blockIdx.x / Mnum;
    int m = blockIdx.x % Mnum;
    __shared__ float sv[256];
    __shared__ int   si[256];
    int t = threadIdx.x;
    float best = -__builtin_inff();
    int bi = t;                               // first covered index
    for (int a = t; a < Anum; a += 256) {
        float v = md[((size_t)b * Anum + a) * Mnum + m];
        if (use_pos && !pos[b * Anum + a]) v = -__builtin_inff();
        if (v > best) { best = v; bi = a; }
    }
    sv[t] = best; si[t] = bi;
    __syncthreads();
    for (int off = 128; off; off >>= 1) {
        if (t < off) {
            float v2 = sv[t + off]; int i2 = si[t + off];
            if (v2 > sv[t] || (v2 == sv[t] && i2 < si[t])) { sv[t] = v2; si[t] = i2; }
        }
        __syncthreads();
    }
    if (t == 0) { outv[blockIdx.x] = sv[0]; outi[blockIdx.x] = si[0]; }
}

// ---- Stage 4: positive anchors + forcing -----------------------------------
__global__ void k_pos(const float* __restrict__ iou_max, int* __restrict__ pos) {
    int idx = blockIdx.x * blockDim.x + threadIdx.x;
    if (idx >= Bsz * Anum) return;
    pos[idx] = (iou_max[idx] >= MD_THRES) ? 1 : 0;
}

__global__ void k_force(const float* __restrict__ max_gt, const int* __restrict__ argmax_gt,
                        int* __restrict__ pos) {
    int t = threadIdx.x;
    if (t >= Bsz * Mnum) return;
    int b = t / Mnum, m = t % Mnum;
    if (max_gt[b * Mnum + m] < MD_THRES)
        pos[b * Anum + argmax_gt[b * Mnum + m]] = 1;   // idempotent store
}

// ---- Stage 5: mw_max per anchor --------------------------------------------
__global__ void k_mw(const float* __restrict__ md, const int* __restrict__ pos,
                     const float* __restrict__ max_pos, const int* __restrict__ arg_pos,
                     float* __restrict__ mw) {
    int idx = blockIdx.x * blockDim.x + threadIdx.x;
    if (idx >= Bsz * Anum) return;
    int b = idx / Anum, a = idx % Anum;
    int p = pos[idx];
    const float* row = md + (size_t)idx * Mnum;
    float mx = -__builtin_inff();
    for (int m = 0; m < Mnum; ++m) {
        float v = row[m];
        bool pm = ((v >= MD_THRES) && p) || (a == arg_pos[b * Mnum + m]);
        float comp = pm ? (1.0f - max_pos[b * Mnum + m]) : v;
        mx = fmaxf(mx, comp + v);
    }
    mw[idx] = mx;
}

// ---- Stage 6: fused focal cls + smooth-L1 reg + pos count ------------------
__global__ void __launch_bounds__(128)
k_loss(const float* __restrict__ cls, const float* __restrict__ reg,
       const float* __restrict__ anc, const float* __restrict__ ann,
       const float* __restrict__ iou_max, const int* __restrict__ iou_arg,
       const int* __restrict__ pos, const float* __restrict__ mw,
       float* __restrict__ cls_part, float* __restrict__ reg_part,
       float* __restrict__ pos_part) {
    int b   = blockIdx.x / NBLK;
    int blk = blockIdx.x % NBLK;
    int t = threadIdx.x;
    int a = blk * 128 + t;
    size_t ia = (size_t)b * Anum + a;
    __shared__ float sc[128], sr[128], sp[128];

    float closs = 0.f, rloss = 0.f, pn = 0.f;
    {
        int   p    = pos[ia];
        float im   = iou_max[ia];
        int   am   = iou_arg[ia];
        const float* g = ann + ((size_t)b * Mnum + am) * 6;
        int   alab = (int)g[5];
        float mwa  = mw[ia];
        const float* crow = cls + ia * Cnum;
        __builtin_prefetch(crow, 0, 1);                // global_prefetch_b8
        for (int c = 0; c < Cnum; ++c) {
            float cv = fminf(fmaxf(crow[c], 1e-4f), 1.f - 1e-4f);
            bool  oh = (alab == c);
            float tgt = (im < MD_THRES - 0.1f) ? 0.f : -1.f;
            if (p) tgt = 0.f;
            if (p && oh) tgt = 1.f;
            if (tgt != -1.f) {
                float af  = (tgt == 1.f) ? 0.25f : 0.75f;
                float x   = (tgt == 1.f) ? (1.f - cv) : cv;
                float fw  = af * x * x;
                float bce = -(tgt * logf(cv + 1e-6f) + (1.f - tgt) * logf(1.f - cv + 1e-6f));
                float sw  = (tgt == 0.f) ? 1.f : 0.f;
                if (p && oh) sw = mwa + 1.f;
                closs += fw * bce * sw;
            }
        }
        pn = p ? 1.f : 0.f;
        if (p) {
            const float* ax = anc + ia * 5;
            const float* rg = reg + ia * 5;
            float gw = fmaxf(g[2], 1.f), gh = fmaxf(g[3], 1.f);
            float tq[5];
            tq[0] = (g[0] - ax[0]) / ax[2];
            tq[1] = (g[1] - ax[1]) / ax[3];
            tq[2] = logf(gw / ax[2]);
            tq[3] = logf(gh / ax[3]);
            tq[4] = tanf(g[4] * DEG2RAD) - tanf(ax[4] * DEG2RAD);
            for (int k = 0; k < 5; ++k) {
                float d = fabsf(rg[k] - tq[k]);
                float s = (d < BETA) ? (0.5f * d * d / BETA) : (d - 0.5f * BETA);
                rloss += s * mwa;
            }
        }
    }
    sc[t] = closs; sr[t] = rloss; sp[t] = pn;
    __syncthreads();
    for (int off = 64; off; off >>= 1) {
        if (t < off) { sc[t] += sc[t + off]; sr[t] += sr[t + off]; sp[t] += sp[t + off]; }
        __syncthreads();
    }
    if (t == 0) {
        cls_part[blockIdx.x] = sc[0];
        reg_part[blockIdx.x] = sr[0];
        pos_part[blockIdx.x] = sp[0];
    }
}

// ---- Stage 7: WMMA-based exact reduction + final scalars -------------------
typedef float v2f __attribute__((ext_vector_type(2)));
typedef float v8f __attribute__((ext_vector_type(8)));

// Sum of NBLK (=128) partials with V_WMMA_F32_16X16X4_F32, B = all-ones.
// D[i][j] = sum_k A[i][k] (replicated over 16 cols) + C, so grand total =
// sum(D) / 16. All-ones B makes the A lane layout irrelevant for the sum.
__device__ float wmma_sum128(const float* __restrict__ p, int lane) {
    v8f acc = {0.f, 0.f, 0.f, 0.f, 0.f, 0.f, 0.f, 0.f};
    v2f bm;  bm[0] = 1.f; bm[1] = 1.f;
    int mrow = lane & 15;
    int koff = (lane < 16) ? 0 : 2;
    for (int it = 0; it < NBLK / 64; ++it) {
        const float* q = p + it * 64;
        v2f av;
        av[0] = q[mrow * 4 + koff];
        av[1] = q[mrow * 4 + koff + 1];
        acc = __builtin_amdgcn_wmma_f32_16x16x4_f32(
            /*neg_a=*/false, av, /*neg_b=*/false, bm,
            /*c_mod=*/(short)0, acc, /*reuse_a=*/false, /*reuse_b=*/false);
    }
    float s = acc[0] + acc[1] + acc[2] + acc[3] + acc[4] + acc[5] + acc[6] + acc[7];
    for (int off = 16; off; off >>= 1) s += __shfl_xor(s, off);
    return s * (1.0f / 16.0f);
}

__global__ void __launch_bounds__(32)
k_final(const float* __restrict__ cls_part, const float* __restrict__ reg_part,
        const float* __restrict__ pos_part, float* __restrict__ out) {
    int lane = threadIdx.x;
    float cl0 = wmma_sum128(cls_part,        lane);
    float cl1 = wmma_sum128(cls_part + NBLK, lane);
    float rl0 = wmma_sum128(reg_part,        lane);
    float rl1 = wmma_sum128(reg_part + NBLK, lane);
    float np0 = wmma_sum128(pos_part,        lane);
    float np1 = wmma_sum128(pos_part + NBLK, lane);
    if (lane == 0) {
        np0 = rintf(np0);
        np1 = rintf(np1);
        float c0 = cl0 / fmaxf(np0, 1.f);
        float c1 = cl1 / fmaxf(np1, 1.f);
        float r0 = (np0 > 0.f) ? rl0 / fmaxf(np0 * 5.f, 1.f) : 0.f;
        float r1 = (np1 > 0.f) ? rl1 / fmaxf(np1 * 5.f, 1.f) : 0.f;
        out[0] = 0.5f * (c0 + c1);   // loss_cls
        out[1] = 0.5f * (r0 + r1);   // loss_reg
    }
}

// ---------------------------------------------------------------------------
extern "C" void kernel_launch(void* const* d_in, const int* in_sizes, int n_in,
                              void* d_out, int out_size, void* d_ws, size_t ws_size,
                              hipStream_t stream) {
    const float* cls  = (const float*)d_in[0];   // (B, A, C)
    const float* reg  = (const float*)d_in[1];   // (B, A, 5)
    const float* anc  = (const float*)d_in[2];   // (B, A, 5)
    const float* ranc = (const float*)d_in[3];   // (B, A, 5)
    const float* ann  = (const float*)d_in[4];   // (B, M, 6)
    float* out = (float*)d_out;

    char* w = (char*)d_ws;
    auto carve = [&](size_t bytes) -> void* {
        void* p = (void*)w;
        w += (bytes + 255) & ~(size_t)255;
        return p;
    };
    float* md    = (float*)carve(sizeof(float) * Bsz * Anum * Mnum);  // 4 MB
    float* ioum  = (float*)carve(sizeof(float) * Bsz * Anum);
    int*   ioua  = (int*)  carve(sizeof(int)   * Bsz * Anum);
    int*   pos   = (int*)  carve(sizeof(int)   * Bsz * Anum);
    float* mw    = (float*)carve(sizeof(float) * Bsz * Anum);
    float* mgt   = (float*)carve(sizeof(float) * Bsz * Mnum);
    int*   agt   = (int*)  carve(sizeof(int)   * Bsz * Mnum);
    float* mpos  = (float*)carve(sizeof(float) * Bsz * Mnum);
    int*   apos  = (int*)  carve(sizeof(int)   * Bsz * Mnum);
    float* cpart = (float*)carve(sizeof(float) * Bsz * NBLK);
    float* rpart = (float*)carve(sizeof(float) * Bsz * NBLK);
    float* ppart = (float*)carve(sizeof(float) * Bsz * NBLK);

    k_md        <<<(Bsz * Anum * Mnum + 255) / 256, 256, 0, stream>>>(anc, ranc, ann, md);
    k_anchor_max<<<(Bsz * Anum + 255) / 256,        256, 0, stream>>>(md, ioum, ioua);
    k_gt_max    <<<Bsz * Mnum,                      256, 0, stream>>>(md, nullptr, 0, mgt, agt);
    k_pos       <<<(Bsz * Anum + 255) / 256,        256, 0, stream>>>(ioum, pos);
    k_force     <<<1, Bsz * Mnum,                        0, stream>>>(mgt, agt, pos);
    k_gt_max    <<<Bsz * Mnum,                      256, 0, stream>>>(md, pos, 1, mpos, apos);
    k_mw        <<<(Bsz * Anum + 255) / 256,        256, 0, stream>>>(md, pos, mpos, apos, mw);
    k_loss      <<<Bsz * NBLK,                      128, 0, stream>>>(cls, reg, anc, ann,
                                                                      ioum, ioua, pos, mw,
                                                                      cpart, rpart, ppart);
    k_final     <<<1, 32,                                0, stream>>>(cpart, rpart, ppart, out);
}